// WindowAttention_27023934226485
// MI455X (gfx1250) — compile-verified
//
#include <hip/hip_runtime.h>

// ---------------------------------------------------------------------------
// Window attention (point-transformer style) for MI455X / gfx1250, wave32.
//   N=50000 pts, K=32 nbrs (== wave32!), C=96, H=6, HC=16, fp32 end to end.
//   GEMMs via V_WMMA_F32_16X16X4_F32; edge phase one-wave-per-point with
//   shfl_xor butterflies for the segment softmax / segment sum.
// ---------------------------------------------------------------------------

typedef float v2f __attribute__((ext_vector_type(2)));
typedef float v8f __attribute__((ext_vector_type(8)));

#define CCH   96
#define HH    6
#define HCH   16
#define KNBR  32
#define WS_F  0.1f
#define QS_F  0.01f

// ---------------------------------------------------------------------------
// Kernel 1: qkv = feats @ w_qkv + b_qkv, split into q (scaled by hc^-0.5=0.25),
// k, v planes. One wave per 16x16 output tile; K loop = 24 WMMA f32 16x16x4.
// A-frag (16x4 f32): lane l -> m=l&15, khalf=l>>4; VGPR j holds K = 2*khalf+j.
// B-frag (4x16 f32): lane l -> n=l&15, khalf=l>>4; VGPR j holds K = 2*khalf+j.
// C/D (16x16 f32):   VGPR r -> row r + 8*khalf, col = l&15.
// ---------------------------------------------------------------------------
__global__ __launch_bounds__(256)
void qkv_gemm_kernel(const float* __restrict__ feats,
                     const float* __restrict__ w_qkv,
                     const float* __restrict__ b_qkv,
                     float* __restrict__ qp,
                     float* __restrict__ kp,
                     float* __restrict__ vp,
                     int n)
{
    const int lane  = threadIdx.x & 31;
    const int wave  = blockIdx.x * 8 + (threadIdx.x >> 5);
    const int ntn   = (3 * CCH) / 16;            // 18 column tiles
    const int ntm   = (n + 15) / 16;
    if (wave >= ntm * ntn) return;               // wave-uniform: EXEC stays all-1s
    const int mt    = wave / ntn;
    const int nt    = wave % ntn;

    const int mrow  = mt * 16 + (lane & 15);
    const int khalf = lane >> 4;
    const int ncol  = nt * 16 + (lane & 15);

    const float* arow = feats + (size_t)mrow * CCH;
    v8f acc = {0.f, 0.f, 0.f, 0.f, 0.f, 0.f, 0.f, 0.f};

    #pragma unroll
    for (int kb = 0; kb < CCH; kb += 4) {
        v2f a, b;
        a[0] = arow[kb + 2 * khalf + 0];
        a[1] = arow[kb + 2 * khalf + 1];
        b[0] = w_qkv[(size_t)(kb + 2 * khalf + 0) * (3 * CCH) + ncol];
        b[1] = w_qkv[(size_t)(kb + 2 * khalf + 1) * (3 * CCH) + ncol];
        acc = __builtin_amdgcn_wmma_f32_16x16x4_f32(false, a, false, b,
                                                    (short)0, acc, false, false);
    }

    const float bias  = b_qkv[ncol];
    const int   sec   = nt / (CCH / 16);         // 0=q, 1=k, 2=v (6 tiles each)
    const int   ccol  = ncol - sec * CCH;
    float*      outp  = (sec == 0) ? qp : ((sec == 1) ? kp : vp);
    const float scale = (sec == 0) ? 0.25f : 1.0f;   // hc^-0.5 = 16^-0.5

    #pragma unroll
    for (int r = 0; r < 8; ++r) {
        const int row = mt * 16 + r + 8 * khalf;
        if (row < n)
            outp[(size_t)row * CCH + ccol] = (acc[r] + bias) * scale;
    }
}

// ---------------------------------------------------------------------------
// Wave-wide butterflies (wave32): every lane ends with the full reduction.
// ---------------------------------------------------------------------------
__device__ __forceinline__ float wave_max(float x) {
    #pragma unroll
    for (int off = 16; off > 0; off >>= 1)
        x = fmaxf(x, __shfl_xor(x, off, 32));
    return x;
}
__device__ __forceinline__ float wave_sum(float x) {
    #pragma unroll
    for (int off = 16; off > 0; off >>= 1)
        x += __shfl_xor(x, off, 32);
    return x;
}

// ---------------------------------------------------------------------------
// Kernel 2: attention. One wave per point i; lane j owns neighbor i*32+j.
// Segment softmax over K=32 == the wave => pure shfl_xor, no LDS/atomics.
// rel tables (40x6x16x3 floats, 46 KB each) stay hot in WGP$/L2.
// ---------------------------------------------------------------------------
__global__ __launch_bounds__(128)
void attn_kernel(const float* __restrict__ qp,
                 const float* __restrict__ kp,
                 const float* __restrict__ vp,
                 const float* __restrict__ coords,
                 const int*   __restrict__ index_1,
                 const float* __restrict__ rqt,
                 const float* __restrict__ rkt,
                 const float* __restrict__ rvt,
                 float* __restrict__ x,
                 int n)
{
    const int lane = threadIdx.x & 31;
    const int i    = blockIdx.x * 4 + (threadIdx.x >> 5);
    if (i >= n) return;                          // wave-uniform

    const int n1 = index_1[i * KNBR + lane];

    // Quantized relative-position table indices (jnp.round == RNE == rintf).
    int rpi[3];
    #pragma unroll
    for (int d = 0; d < 3; ++d) {
        float rel = coords[i * 3 + d] - coords[n1 * 3 + d];
        rel = rintf(rel * 100000.0f) / 100000.0f;
        rpi[d] = (int)floorf((rel + 2.0f * WS_F - 0.0001f) / QS_F);
    }

    const float* qrow = qp + (size_t)i  * CCH;   // broadcast across the wave
    const float* krow = kp + (size_t)n1 * CCH;   // per-lane gather (L2-resident)

    // Logits with fused rel-pos bias: a = q·(k + Σd rq[d]) + k·(Σd rk[d])
    float attn[HH];
    #pragma unroll
    for (int h = 0; h < HH; ++h) {
        float a = 0.f;
        #pragma unroll
        for (int c = 0; c < HCH; ++c) {
            const float qv = qrow[h * HCH + c];
            const float kv = krow[h * HCH + c];
            float rq = 0.f, rk = 0.f;
            #pragma unroll
            for (int d = 0; d < 3; ++d) {
                const int tb = ((rpi[d] * HH + h) * HCH + c) * 3 + d;
                rq += rqt[tb];
                rk += rkt[tb];
            }
            a = fmaf(qv, kv + rq, fmaf(kv, rk, a));
        }
        attn[h] = a;
    }

    // Segment softmax across the 32 neighbors == the 32 lanes.
    float p[HH];
    #pragma unroll
    for (int h = 0; h < HH; ++h) {
        const float mx = wave_max(attn[h]);
        const float e  = expf(attn[h] - mx);
        p[h] = e / wave_sum(e);
    }

    // x[i,h,c] = Σ_lanes p[h] * (v[n1,h,c] + Σd rvt[rpi[d],h,c,d])
    const float* vrow = vp + (size_t)n1 * CCH;
    #pragma unroll
    for (int h = 0; h < HH; ++h) {
        #pragma unroll
        for (int c = 0; c < HCH; ++c) {
            float tv = 0.f;
            #pragma unroll
            for (int d = 0; d < 3; ++d)
                tv += rvt[((rpi[d] * HH + h) * HCH + c) * 3 + d];
            float val = p[h] * (vrow[h * HCH + c] + tv);
            val = wave_sum(val);
            if (lane == ((h * HCH + c) & 31))
                x[(size_t)i * CCH + h * HCH + c] = val;
        }
    }
}

// ---------------------------------------------------------------------------
// Kernel 3: out = x @ w_proj + b_proj (96x96). Same WMMA tiling as kernel 1.
// ---------------------------------------------------------------------------
__global__ __launch_bounds__(256)
void proj_gemm_kernel(const float* __restrict__ x,
                      const float* __restrict__ w_proj,
                      const float* __restrict__ b_proj,
                      float* __restrict__ out,
                      int n)
{
    const int lane  = threadIdx.x & 31;
    const int wave  = blockIdx.x * 8 + (threadIdx.x >> 5);
    const int ntn   = CCH / 16;                  // 6 column tiles
    const int ntm   = (n + 15) / 16;
    if (wave >= ntm * ntn) return;
    const int mt    = wave / ntn;
    const int nt    = wave % ntn;

    const int mrow  = mt * 16 + (lane & 15);
    const int khalf = lane >> 4;
    const int ncol  = nt * 16 + (lane & 15);

    const float* arow = x + (size_t)mrow * CCH;
    v8f acc = {0.f, 0.f, 0.f, 0.f, 0.f, 0.f, 0.f, 0.f};

    #pragma unroll
    for (int kb = 0; kb < CCH; kb += 4) {
        v2f a, b;
        a[0] = arow[kb + 2 * khalf + 0];
        a[1] = arow[kb + 2 * khalf + 1];
        b[0] = w_proj[(size_t)(kb + 2 * khalf + 0) * CCH + ncol];
        b[1] = w_proj[(size_t)(kb + 2 * khalf + 1) * CCH + ncol];
        acc = __builtin_amdgcn_wmma_f32_16x16x4_f32(false, a, false, b,
                                                    (short)0, acc, false, false);
    }

    const float bias = b_proj[ncol];
    #pragma unroll
    for (int r = 0; r < 8; ++r) {
        const int row = mt * 16 + r + 8 * khalf;
        if (row < n)
            out[(size_t)row * CCH + ncol] = acc[r] + bias;
    }
}

// ---------------------------------------------------------------------------
extern "C" void kernel_launch(void* const* d_in, const int* in_sizes, int n_in,
                              void* d_out, int out_size, void* d_ws, size_t ws_size,
                              hipStream_t stream)
{
    const float* feats   = (const float*)d_in[0];
    const float* coords  = (const float*)d_in[1];
    /* d_in[2] index_0: implied by contiguous K=32 segments */
    const int*   index_1 = (const int*)  d_in[3];
    /* d_in[4] index_0_offsets, d_in[5] n_max: unused (uniform K) */
    const float* w_qkv   = (const float*)d_in[6];
    const float* b_qkv   = (const float*)d_in[7];
    const float* rqt     = (const float*)d_in[8];
    const float* rkt     = (const float*)d_in[9];
    const float* rvt     = (const float*)d_in[10];
    const float* w_proj  = (const float*)d_in[11];
    const float* b_proj  = (const float*)d_in[12];
    float*       out     = (float*)d_out;

    const int n = in_sizes[0] / CCH;             // 50000

    // Workspace planes: q | k | v  (x aliases q: q rows are consumed only by
    // their owning wave before that wave writes its x row).
    float* qp = (float*)d_ws;
    float* kp = qp + (size_t)n * CCH;
    float* vp = kp + (size_t)n * CCH;
    float* xp = qp;

    const int qkv_tiles  = ((n + 15) / 16) * 18;
    qkv_gemm_kernel<<<(qkv_tiles + 7) / 8, 256, 0, stream>>>(
        feats, w_qkv, b_qkv, qp, kp, vp, n);

    attn_kernel<<<(n + 3) / 4, 128, 0, stream>>>(
        qp, kp, vp, coords, index_1, rqt, rkt, rvt, xp, n);

    const int proj_tiles = ((n + 15) / 16) * 6;
    proj_gemm_kernel<<<(proj_tiles + 7) / 8, 256, 0, stream>>>(
        xp, w_proj, b_proj, out, n);
}